// GNNModel_31267361915355
// MI455X (gfx1250) — compile-verified
//
#include <hip/hip_runtime.h>

typedef __attribute__((ext_vector_type(2))) float v2f;
typedef __attribute__((ext_vector_type(8))) float v8f;

#define NEG_SLOPE 0.2f

// ---------------------------------------------------------------------------
// Utility: fill a float array
// ---------------------------------------------------------------------------
__global__ void fill_f32(float* __restrict__ p, float v, long long n) {
  long long i = (long long)blockIdx.x * blockDim.x + threadIdx.x;
  if (i < n) p[i] = v;
}

// ---------------------------------------------------------------------------
// fp32 WMMA GEMM: H[N x COLS] = X[N x K] @ W[K x COLS]
// One wave (32 lanes) per 16-row tile. COLS is 16 or 32.
// A-frag (16x4 f32): lanes 0-15 hold {K=k,k+1} of row M=lane; lanes 16-31 hold
// {K=k+2,k+3} of row M=lane-16. B-frag symmetric (4x16). C/D: vgpr j, lane<16
// -> [M=j][N=lane]; lane>=16 -> [M=j+8][N=lane-16].
// Requires nrows % 16 == 0 (EXEC must be all ones for WMMA).
// ---------------------------------------------------------------------------
template <int K, int COLS>
__global__ void gemm_wmma_f32(const float* __restrict__ X,
                              const float* __restrict__ W,
                              float* __restrict__ H) {
  const int lane = threadIdx.x & 31;
  const int half = lane >> 4;   // 0 or 1
  const int l16  = lane & 15;
  const int tile = blockIdx.x;
  const int row  = tile * 16 + l16;

  v8f c0 = {};
  v8f c1 = {};

#pragma unroll
  for (int k = 0; k < K; k += 4) {
    const int kk = k + half * 2;
    v2f a  = { X[row * K + kk], X[row * K + kk + 1] };
    v2f b0 = { W[kk * COLS + l16], W[(kk + 1) * COLS + l16] };
    c0 = __builtin_amdgcn_wmma_f32_16x16x4_f32(false, a, false, b0,
                                               (short)0, c0, false, false);
    if (COLS == 32) {
      v2f b1 = { W[kk * COLS + 16 + l16], W[(kk + 1) * COLS + 16 + l16] };
      c1 = __builtin_amdgcn_wmma_f32_16x16x4_f32(false, a, false, b1,
                                                 (short)0, c1, false, false);
    }
  }

#pragma unroll
  for (int j = 0; j < 8; ++j) {
    const int m = j + half * 8;
    H[(tile * 16 + m) * COLS + l16] = c0[j];
    if (COLS == 32) H[(tile * 16 + m) * COLS + 16 + l16] = c1[j];
  }
}

// ---------------------------------------------------------------------------
// Per-node attention logits: aS[i] = h[i,:]·att_src ; aD[i] = h[i,:]·att_dst
// ---------------------------------------------------------------------------
template <int F>
__global__ void attn_logits(const float* __restrict__ H,
                            const float* __restrict__ att_s,
                            const float* __restrict__ att_d,
                            float* __restrict__ aS, float* __restrict__ aD,
                            int n) {
  int i = blockIdx.x * blockDim.x + threadIdx.x;
  if (i >= n) return;
  float s = 0.f, d = 0.f;
#pragma unroll
  for (int c = 0; c < F; ++c) {
    float h = H[i * F + c];
    s += h * att_s[c];
    d += h * att_d[c];
  }
  aS[i] = s;
  aD[i] = d;
}

// ---------------------------------------------------------------------------
// Edge helpers
// ---------------------------------------------------------------------------
__device__ __forceinline__ void edge_sd(const int* __restrict__ ei, int E,
                                        int eid, int& s, int& d) {
  if (eid < E) { s = ei[eid]; d = ei[E + eid]; }
  else         { s = eid - E; d = eid - E; }   // self loops appended
}

__device__ __forceinline__ float lrelu(float e) {
  return e > 0.f ? e : NEG_SLOPE * e;
}

// sign-aware float atomic max via int punning (init must be -inf)
__device__ __forceinline__ void atomicMaxF(float* addr, float val) {
  if (val >= 0.f)
    atomicMax((int*)addr, __float_as_int(val));
  else
    atomicMin((unsigned int*)addr, __float_as_uint(val));
}

// Pass 1: per-dst segment max of leaky_relu(aS[src]+aD[dst])
__global__ void edge_max(const int* __restrict__ ei, int E, int ET,
                         const float* __restrict__ aS,
                         const float* __restrict__ aD,
                         float* __restrict__ m) {
  int eid = blockIdx.x * blockDim.x + threadIdx.x;
  if (eid >= ET) return;
  int s, d;
  edge_sd(ei, E, eid, s, d);
  atomicMaxF(&m[d], lrelu(aS[s] + aD[d]));
}

// Pass 2: per-dst sum of exp(e - m[dst])
__global__ void edge_denom(const int* __restrict__ ei, int E, int ET,
                           const float* __restrict__ aS,
                           const float* __restrict__ aD,
                           const float* __restrict__ m,
                           float* __restrict__ denom) {
  int eid = blockIdx.x * blockDim.x + threadIdx.x;
  if (eid >= ET) return;
  int s, d;
  edge_sd(ei, E, eid, s, d);
  float e = lrelu(aS[s] + aD[d]);
  atomicAdd(&denom[d], __expf(e - m[d]));
}

// Pass 3: out[dst,:] += alpha * h[src,:]  (F lanes cooperate per edge)
template <int F>
__global__ void edge_scatter(const int* __restrict__ ei, int E, int ET,
                             const float* __restrict__ aS,
                             const float* __restrict__ aD,
                             const float* __restrict__ m,
                             const float* __restrict__ denom,
                             const float* __restrict__ H,
                             float* __restrict__ out) {
  long long t = (long long)blockIdx.x * blockDim.x + threadIdx.x;
  int eid = (int)(t / F);
  int c   = (int)(t & (F - 1));
  if (eid >= ET) return;
  int s, d;
  edge_sd(ei, E, eid, s, d);
  float e     = lrelu(aS[s] + aD[d]);
  float alpha = __expf(e - m[d]) / denom[d];
  atomicAdd(&out[(long long)d * F + c], H[(long long)s * F + c] * alpha);
}

// Bias + ReLU in place: out[i*F+c] = relu(out[i*F+c] + b[c])
template <int F>
__global__ void bias_relu(float* __restrict__ out, const float* __restrict__ b,
                          long long total) {
  long long t = (long long)blockIdx.x * blockDim.x + threadIdx.x;
  if (t >= total) return;
  int c = (int)(t & (F - 1));
  float v = out[t] + b[c];
  out[t] = v > 0.f ? v : 0.f;
}

// Final: h = relu(out2 + b2) -> d_out[0 : N*16]; sigmoid(h@Wc+bc) -> d_out[N*16:]
__global__ void finalize(const float* __restrict__ out2,
                         const float* __restrict__ b2,
                         const float* __restrict__ Wc,
                         const float* __restrict__ bc,
                         float* __restrict__ hout,
                         float* __restrict__ scores, int n) {
  int i = blockIdx.x * blockDim.x + threadIdx.x;
  if (i >= n) return;
  float acc = bc[0];
#pragma unroll
  for (int c = 0; c < 16; ++c) {
    float v = out2[i * 16 + c] + b2[c];
    v = v > 0.f ? v : 0.f;
    hout[i * 16 + c] = v;
    acc += v * Wc[c];
  }
  scores[i] = 1.f / (1.f + __expf(-acc));
}

// ---------------------------------------------------------------------------
// Launch
// ---------------------------------------------------------------------------
extern "C" void kernel_launch(void* const* d_in, const int* in_sizes, int n_in,
                              void* d_out, int out_size, void* d_ws,
                              size_t ws_size, hipStream_t stream) {
  const float* x     = (const float*)d_in[0];
  const int*   ei    = (const int*)d_in[1];
  const float* W1    = (const float*)d_in[2];
  const float* atS1  = (const float*)d_in[3];
  const float* atD1  = (const float*)d_in[4];
  const float* b1    = (const float*)d_in[5];
  const float* W2    = (const float*)d_in[6];
  const float* atS2  = (const float*)d_in[7];
  const float* atD2  = (const float*)d_in[8];
  const float* b2    = (const float*)d_in[9];
  const float* Wc    = (const float*)d_in[10];
  const float* bc    = (const float*)d_in[11];

  const int n  = in_sizes[0] / 128;   // 50000 (multiple of 16)
  const int E  = in_sizes[1] / 2;     // 1600000
  const int ET = E + n;               // with self loops

  // workspace carve-up (floats)
  float* p    = (float*)d_ws;
  float* h1   = p; p += (long long)n * 32;
  float* out1 = p; p += (long long)n * 32;   // becomes relu(h1_agg+b1)
  float* h2   = p; p += (long long)n * 16;
  float* out2 = p; p += (long long)n * 16;
  float* aS1  = p; p += n;
  float* aD1  = p; p += n;
  float* m1   = p; p += n;
  float* dn1  = p; p += n;
  float* aS2  = p; p += n;
  float* aD2  = p; p += n;
  float* m2   = p; p += n;
  float* dn2  = p; p += n;

  const int B = 256;
  auto blocks = [](long long work, int bs) { return (int)((work + bs - 1) / bs); };
  const int tiles = n / 16;

  // ---- Layer 1 ----
  fill_f32<<<blocks(n, B), B, 0, stream>>>(m1, -INFINITY, n);
  fill_f32<<<blocks(n, B), B, 0, stream>>>(dn1, 0.f, n);
  fill_f32<<<blocks((long long)n * 32, B), B, 0, stream>>>(out1, 0.f, (long long)n * 32);

  gemm_wmma_f32<128, 32><<<tiles, 32, 0, stream>>>(x, W1, h1);
  attn_logits<32><<<blocks(n, B), B, 0, stream>>>(h1, atS1, atD1, aS1, aD1, n);

  edge_max<<<blocks(ET, B), B, 0, stream>>>(ei, E, ET, aS1, aD1, m1);
  edge_denom<<<blocks(ET, B), B, 0, stream>>>(ei, E, ET, aS1, aD1, m1, dn1);
  edge_scatter<32><<<blocks((long long)ET * 32, B), B, 0, stream>>>(
      ei, E, ET, aS1, aD1, m1, dn1, h1, out1);
  bias_relu<32><<<blocks((long long)n * 32, B), B, 0, stream>>>(out1, b1,
                                                               (long long)n * 32);

  // ---- Layer 2 ----
  fill_f32<<<blocks(n, B), B, 0, stream>>>(m2, -INFINITY, n);
  fill_f32<<<blocks(n, B), B, 0, stream>>>(dn2, 0.f, n);
  fill_f32<<<blocks((long long)n * 16, B), B, 0, stream>>>(out2, 0.f, (long long)n * 16);

  gemm_wmma_f32<32, 16><<<tiles, 32, 0, stream>>>(out1, W2, h2);
  attn_logits<16><<<blocks(n, B), B, 0, stream>>>(h2, atS2, atD2, aS2, aD2, n);

  edge_max<<<blocks(ET, B), B, 0, stream>>>(ei, E, ET, aS2, aD2, m2);
  edge_denom<<<blocks(ET, B), B, 0, stream>>>(ei, E, ET, aS2, aD2, m2, dn2);
  edge_scatter<16><<<blocks((long long)ET * 16, B), B, 0, stream>>>(
      ei, E, ET, aS2, aD2, m2, dn2, h2, out2);

  // ---- Finalize: h -> d_out[0:N*16], scores -> d_out[N*16:N*17] ----
  float* hout   = (float*)d_out;
  float* scores = hout + (long long)n * 16;
  finalize<<<blocks(n, B), B, 0, stream>>>(out2, b2, Wc, bc, hout, scores, n);
}